// OptimalModule_38036230373754
// MI455X (gfx1250) — compile-verified
//
#include <hip/hip_runtime.h>
#include <hip/hip_bf16.h>
#include <math.h>

#define AUXN   9
#define REFN   10001
#define XCOLS  10011
#define BATCH  512
#define LDA    68            // padded LDS row stride (floats) to dodge bank conflicts

typedef __attribute__((ext_vector_type(2))) float v2f;
typedef __attribute__((ext_vector_type(8))) float v8f;

__device__ __forceinline__ float leaky(float v) { return v > 0.0f ? v : 0.01f * v; }

// ---------------------------------------------------------------------------
// Kernel 1: per-batch-row attention scan (VALU-bound part).
// One block = one batch row b. 256 threads = 8 waves.
//   Phase A (32 lanes): hidden, q, u = Wk2*q (scaled), c = q.bk2 (scaled)
//   Pass 1 (256 thr, thread-per-n): scores -> LDS, block max
//   Pass 2 (8 waves, lane-per-j): online weighted accumulation of leaky(t*wv1+bv1)
//   Epilogue (32 lanes): att_out = (S/L) @ Wv2 + bv2 ; emit h64[b] = [hidden | att]
// ---------------------------------------------------------------------------
__global__ void __launch_bounds__(256)
attn_scan_kernel(const float* __restrict__ x,
                 const float* __restrict__ Ws1, const float* __restrict__ bs1,
                 const float* __restrict__ Ws2, const float* __restrict__ bs2,
                 const float* __restrict__ Wq1, const float* __restrict__ bq1,
                 const float* __restrict__ Wq2, const float* __restrict__ bq2,
                 const float* __restrict__ Wk1, const float* __restrict__ bk1,
                 const float* __restrict__ Wk2, const float* __restrict__ bk2,
                 const float* __restrict__ Wv1, const float* __restrict__ bv1,
                 const float* __restrict__ Wv2, const float* __restrict__ bv2,
                 float* __restrict__ h64)
{
    __shared__ float s_scores[REFN + 15];
    __shared__ float s_hidden[32];
    __shared__ float s_tmp[64];            // staging: h1 / q1, then q in [32..63]
    __shared__ float s_wk1[32], s_bk1[32], s_u[32];
    __shared__ float s_wv1[32], s_bv1[32];
    __shared__ float s_red[8 * 33];        // per-wave partial S_j
    __shared__ float s_lw[8];              // per-wave partial L
    __shared__ float s_c, s_max, s_L;

    const int b    = blockIdx.x;
    const int tid  = threadIdx.x;
    const int wave = tid >> 5;
    const int lane = tid & 31;
    const float* __restrict__ xrow = x + (size_t)b * XCOLS;
    const float inv_sqrt_d = 0.17677669529663687f;   // 1/sqrt(32)

    // ---------------- Phase A: tiny per-row MLP (front-end + query) ----------
    if (tid < 32) {                                   // h1 = leaky(aux@Ws1+bs1)
        float acc = bs1[tid];
        #pragma unroll
        for (int i = 0; i < AUXN; ++i) acc = fmaf(xrow[i], Ws1[i * 32 + tid], acc);
        s_tmp[tid] = leaky(acc);
    }
    __syncthreads();
    if (tid < 32) {                                   // hidden = leaky(h1@Ws2+bs2)
        float acc = bs2[tid];
        #pragma unroll
        for (int i = 0; i < 32; ++i) acc = fmaf(s_tmp[i], Ws2[i * 32 + tid], acc);
        s_hidden[tid] = leaky(acc);
    }
    __syncthreads();
    if (tid < 32) {                                   // q1 = leaky(hidden@Wq1+bq1)
        float acc = bq1[tid];
        #pragma unroll
        for (int i = 0; i < 32; ++i) acc = fmaf(s_hidden[i], Wq1[i * 32 + tid], acc);
        s_tmp[tid] = leaky(acc);
    }
    __syncthreads();
    if (tid < 32) {                                   // q = q1@Wq2+bq2
        float acc = bq2[tid];
        #pragma unroll
        for (int i = 0; i < 32; ++i) acc = fmaf(s_tmp[i], Wq2[i * 32 + tid], acc);
        s_tmp[32 + tid] = acc;
    }
    __syncthreads();
    if (tid < 32) {                                   // u_j = (Wk2 q)_j / sqrt(d)
        float acc = 0.0f;
        #pragma unroll
        for (int d = 0; d < 32; ++d) acc = fmaf(Wk2[tid * 32 + d], s_tmp[32 + d], acc);
        s_u[tid]   = acc * inv_sqrt_d;
        s_wk1[tid] = Wk1[tid];  s_bk1[tid] = bk1[tid];
        s_wv1[tid] = Wv1[tid];  s_bv1[tid] = bv1[tid];
    }
    if (tid == 0) {                                   // c = q.bk2 / sqrt(d)
        float acc = 0.0f;
        for (int d = 0; d < 32; ++d) acc = fmaf(s_tmp[32 + d], bk2[d], acc);
        s_c = acc * inv_sqrt_d;
    }
    __syncthreads();

    // ---------------- Pass 1: scores (thread-per-n, j fully unrolled in VGPRs)
    float rk[32], rb[32], ru[32];
    #pragma unroll
    for (int j = 0; j < 32; ++j) { rk[j] = s_wk1[j]; rb[j] = s_bk1[j]; ru[j] = s_u[j]; }
    const float cc = s_c;

    float lmax = -INFINITY;
    for (int n = tid; n < REFN; n += 256) {
        float t = xrow[AUXN + 1 + n];
        __builtin_prefetch(xrow + AUXN + 1 + n + 1024, 0, 0);   // global_prefetch_b8
        float sc = cc;
        #pragma unroll
        for (int j = 0; j < 32; ++j) {
            float z = fmaf(t, rk[j], rb[j]);
            z = z > 0.0f ? z : 0.01f * z;
            sc = fmaf(z, ru[j], sc);
        }
        s_scores[n] = sc;
        lmax = fmaxf(lmax, sc);
    }
    #pragma unroll
    for (int off = 16; off > 0; off >>= 1)
        lmax = fmaxf(lmax, __shfl_xor(lmax, off, 32));
    if (lane == 0) s_red[wave] = lmax;
    __syncthreads();
    if (tid == 0) {
        float m = s_red[0];
        #pragma unroll
        for (int w = 1; w < 8; ++w) m = fmaxf(m, s_red[w]);
        s_max = m;
    }
    __syncthreads();
    const float gmax = s_max;

    // ---------------- Pass 2: weighted value accumulation (lane-per-j) -------
    {
        const float rv  = s_wv1[lane];
        const float rbv = s_bv1[lane];
        float sj = 0.0f, l = 0.0f;
        for (int n = wave; n < REFN; n += 8) {
            float t  = xrow[AUXN + 1 + n];               // wave-uniform broadcast load
            float p  = __expf(s_scores[n] - gmax);
            l += p;
            float z = fmaf(t, rv, rbv);
            z = z > 0.0f ? z : 0.01f * z;
            sj = fmaf(p, z, sj);
        }
        s_red[wave * 33 + lane] = sj;
        if (lane == 0) s_lw[wave] = l;
    }
    __syncthreads();

    // ---------------- Epilogue: att_out = (S/L)@Wv2 + bv2 --------------------
    if (tid < 32) {
        float S = 0.0f;
        #pragma unroll
        for (int w = 0; w < 8; ++w) S += s_red[w * 33 + tid];
        s_tmp[tid] = S;
        if (tid == 0) {
            float L = 0.0f;
            #pragma unroll
            for (int w = 0; w < 8; ++w) L += s_lw[w];
            s_L = L;
        }
    }
    __syncthreads();
    if (tid < 32) {
        const float inv = 1.0f / s_L;
        float acc = bv2[tid];
        #pragma unroll
        for (int j = 0; j < 32; ++j) acc = fmaf(s_tmp[j] * inv, Wv2[j * 32 + tid], acc);
        h64[(size_t)b * 64 + tid]      = s_hidden[tid];  // [hidden | att_out]
        h64[(size_t)b * 64 + 32 + tid] = acc;
    }
}

// ---------------------------------------------------------------------------
// Kernel 2: MLP head via V_WMMA_F32_16X16X4_F32.
// One wave (32 threads) per 16-row M-tile; 32 blocks cover 512 rows.
// GEMM chain: (16x64)@Wc1(64x64) -> @Wc2(64x32) -> @Wc3(32x32) -> @Wc4(32x3)
// VGPR layouts per CDNA5 ISA 7.12.2:
//   A 16x4 f32 : lanes 0-15 rows M, v.x=K0 v.y=K1 ; lanes 16-31 v.x=K2 v.y=K3
//   B 4x16 f32 : v.x = row K(+2*half), v.y = row K+1 ; lane&15 = column N
//   C 16x16 f32: c[r] at row r+8*(lane>>4), col lane&15
// ---------------------------------------------------------------------------
__device__ __forceinline__ v8f wmma4(v2f a, v2f b, v8f c) {
    return __builtin_amdgcn_wmma_f32_16x16x4_f32(false, a, false, b,
                                                 (short)0, c, false, false);
}

__device__ __forceinline__ void gemm_layer_lds(const float* __restrict__ sIn, int K,
                                               const float* __restrict__ W, int ldw,
                                               const float* __restrict__ bias,
                                               int nOut, float* __restrict__ sOut,
                                               int lane)
{
    const int half = lane >> 4;
    const int n    = lane & 15;
    for (int n0 = 0; n0 < nOut; n0 += 16) {
        v8f c = {};
        for (int k0 = 0; k0 < K; k0 += 4) {
            v2f a, bm;
            a.x  = sIn[(lane & 15) * LDA + k0 + 2 * half];
            a.y  = sIn[(lane & 15) * LDA + k0 + 2 * half + 1];
            bm.x = W[(k0 + 2 * half)     * ldw + n0 + n];
            bm.y = W[(k0 + 2 * half + 1) * ldw + n0 + n];
            c = wmma4(a, bm, c);
        }
        const float bn = bias[n0 + n];
        #pragma unroll
        for (int r = 0; r < 8; ++r) {
            float v = c[r] + bn;
            v = v > 0.0f ? v : 0.01f * v;                // leaky
            sOut[(r + 8 * half) * LDA + n0 + n] = v;
        }
    }
}

__global__ void __launch_bounds__(32)
mlp_head_kernel(const float* __restrict__ h64,
                const float* __restrict__ Wc1, const float* __restrict__ bc1,
                const float* __restrict__ Wc2, const float* __restrict__ bc2,
                const float* __restrict__ Wc3, const float* __restrict__ bc3,
                const float* __restrict__ Wc4, const float* __restrict__ bc4,
                float* __restrict__ out)
{
    __shared__ float sA[16 * LDA];
    __shared__ float sB[16 * LDA];

    const int lane = threadIdx.x;
    const int m0   = blockIdx.x * 16;
    const int half = lane >> 4;
    const int n    = lane & 15;

    // stage 16x64 activation tile into LDS
    for (int i = lane; i < 16 * 64; i += 32)
        sA[(i >> 6) * LDA + (i & 63)] = h64[(size_t)(m0 + (i >> 6)) * 64 + (i & 63)];
    __syncthreads();

    gemm_layer_lds(sA, 64, Wc1, 64, bc1, 64, sB, lane);   // -> leaky, 16x64
    __syncthreads();
    gemm_layer_lds(sB, 64, Wc2, 32, bc2, 32, sA, lane);   // -> leaky, 16x32
    __syncthreads();
    gemm_layer_lds(sA, 32, Wc3, 32, bc3, 32, sB, lane);   // -> leaky, 16x32
    __syncthreads();

    // final layer: (16x32) @ Wc4(32x3) + bc4, no activation, 3 valid columns
    {
        v8f c = {};
        for (int k0 = 0; k0 < 32; k0 += 4) {
            v2f a, bm;
            a.x  = sB[(lane & 15) * LDA + k0 + 2 * half];
            a.y  = sB[(lane & 15) * LDA + k0 + 2 * half + 1];
            bm.x = (n < 3) ? Wc4[(k0 + 2 * half)     * 3 + n] : 0.0f;
            bm.y = (n < 3) ? Wc4[(k0 + 2 * half + 1) * 3 + n] : 0.0f;
            c = wmma4(a, bm, c);
        }
        if (n < 3) {
            const float bn = bc4[n];
            #pragma unroll
            for (int r = 0; r < 8; ++r)
                out[(size_t)(m0 + r + 8 * half) * 3 + n] = c[r] + bn;
        }
    }
}

// ---------------------------------------------------------------------------
extern "C" void kernel_launch(void* const* d_in, const int* in_sizes, int n_in,
                              void* d_out, int out_size, void* d_ws, size_t ws_size,
                              hipStream_t stream)
{
    const float* x   = (const float*)d_in[0];
    const float* Ws1 = (const float*)d_in[1];
    const float* bs1 = (const float*)d_in[2];
    const float* Ws2 = (const float*)d_in[3];
    const float* bs2 = (const float*)d_in[4];
    const float* Wq1 = (const float*)d_in[5];
    const float* bq1 = (const float*)d_in[6];
    const float* Wq2 = (const float*)d_in[7];
    const float* bq2 = (const float*)d_in[8];
    const float* Wk1 = (const float*)d_in[9];
    const float* bk1 = (const float*)d_in[10];
    const float* Wk2 = (const float*)d_in[11];
    const float* bk2 = (const float*)d_in[12];
    const float* Wv1 = (const float*)d_in[13];
    const float* bv1 = (const float*)d_in[14];
    const float* Wv2 = (const float*)d_in[15];
    const float* bv2 = (const float*)d_in[16];
    const float* Wc1 = (const float*)d_in[17];
    const float* bc1 = (const float*)d_in[18];
    const float* Wc2 = (const float*)d_in[19];
    const float* bc2 = (const float*)d_in[20];
    const float* Wc3 = (const float*)d_in[21];
    const float* bc3 = (const float*)d_in[22];
    const float* Wc4 = (const float*)d_in[23];
    const float* bc4 = (const float*)d_in[24];

    float* h64 = (float*)d_ws;                       // (512, 64) staging

    attn_scan_kernel<<<BATCH, 256, 0, stream>>>(x,
        Ws1, bs1, Ws2, bs2, Wq1, bq1, Wq2, bq2,
        Wk1, bk1, Wk2, bk2, Wv1, bv1, Wv2, bv2, h64);

    mlp_head_kernel<<<BATCH / 16, 32, 0, stream>>>(h64,
        Wc1, bc1, Wc2, bc2, Wc3, bc3, Wc4, bc4, (float*)d_out);
}